// RoIHeads_83631603187920
// MI455X (gfx1250) — compile-verified
//
#include <hip/hip_runtime.h>
#include <stdint.h>

// ---------------- problem constants (match reference) ----------------
#define B_    8
#define N_    8000
#define C_    81
#define CM1   80
#define M_    (N_ * CM1)          // 640000 candidates per image
#define KPRE  4096
#define DET   100
#define IMGW  1216.0f
#define IMGH  800.0f
#define NEGV  (-1e9f)
#define SCORE_TH 0.05f
#define NMS_TH   0.5f
#define MINSZ    0.01f
#define CLIPV    4.135166556742356f   // log(1000/16)

// ---------------- workspace layout (bytes) ----------------
#define OFF_KEYS   ((size_t)0)
#define KEYS_BYTES ((size_t)B_ * M_ * 4)                       // 20,480,000
#define OFF_HIST   (OFF_KEYS + KEYS_BYTES)                     // 8*256*4
#define OFF_PREFIX (OFF_HIST + (size_t)B_ * 256 * 4 + 224)     // align-ish
#define OFF_KCUR   (OFF_PREFIX + 256)
#define OFF_CNTGT  (OFF_KCUR + 256)
#define OFF_CNTEQ  (OFF_CNTGT + 256)
#define OFF_CKEY   (OFF_CNTEQ + 256)
#define OFF_CIDX   (OFF_CKEY + (size_t)B_ * KPRE * 4)
// total ~20.76 MB

// ---------------- helpers ----------------
__device__ __forceinline__ unsigned f2k(float f) {
    unsigned u = __float_as_uint(f);
    return (u & 0x80000000u) ? ~u : (u | 0x80000000u);
}
__device__ __forceinline__ float k2f(unsigned k) {
    unsigned u = (k & 0x80000000u) ? (k & 0x7FFFFFFFu) : ~k;
    return __uint_as_float(u);
}

// gfx1250 async global->LDS copy (ASYNCcnt path). lds_addr is the numeric LDS
// byte address (per-lane), gbl_off a per-lane byte offset vs. the uniform
// 64-bit base in SGPRs (GVS addressing mode).
__device__ __forceinline__ void async_g2l_b128(unsigned lds_addr, unsigned gbl_off,
                                               unsigned long long gbl_base) {
    asm volatile("global_load_async_to_lds_b128 %0, %1, %2"
                 :
                 : "v"(lds_addr), "v"(gbl_off), "s"(gbl_base)
                 : "memory");
}
__device__ __forceinline__ void wait_async0() {
    asm volatile("s_wait_asynccnt 0" ::: "memory");
}

// ---------------- kernel 1: softmax + decode + mask -> sortable keys --------
// one wave (32 lanes) per (b,n) row; block = 256 threads = 8 rows
__global__ __launch_bounds__(256) void k_scores(const float* __restrict__ logits,
                                                const float* __restrict__ rel,
                                                const float* __restrict__ props,
                                                unsigned* __restrict__ keys) {
    int row  = blockIdx.x * 8 + (threadIdx.x >> 5);
    int lane = threadIdx.x & 31;
    if (row >= B_ * N_) return;
    int b = row / N_;
    int n = row % N_;

    const float* lrow = logits + (size_t)row * C_;
    const float* rrow = rel + (size_t)row * (C_ * 4);
    __builtin_prefetch(rrow, 0, 0);   // global_prefetch_b8

    // softmax denominator (max-subtracted), wave32 reductions
    float m = -3.4e38f;
    for (int c = lane; c < C_; c += 32) m = fmaxf(m, lrow[c]);
    for (int o = 16; o; o >>= 1) m = fmaxf(m, __shfl_xor(m, o, 32));

    float s = __expf(lrow[lane] - m);
    if (lane + 32 < C_) s += __expf(lrow[lane + 32] - m);
    if (lane + 64 < C_) s += __expf(lrow[lane + 64] - m);
    for (int o = 16; o; o >>= 1) s += __shfl_xor(s, o, 32);
    float inv = 1.0f / s;

    const float* pr = props + (size_t)row * 4;
    float p0 = pr[0], p1 = pr[1], p2 = pr[2], p3 = pr[3];
    float w = p2 - p0, h = p3 - p1;
    float cx = p0 + 0.5f * w, cy = p1 + 0.5f * h;

    unsigned* krow = keys + (size_t)b * M_ + (size_t)n * CM1;

    for (int i = 0; i < CM1; i += 32) {
        if (lane + i < CM1) {
            int c = 1 + lane + i;                        // classes 1..80
            float prob = __expf(lrow[c] - m) * inv;
            const float* rv = rrow + c * 4;
            float dx = rv[0] * 0.1f, dy = rv[1] * 0.1f;
            float dw = fminf(rv[2] * 0.2f, CLIPV);
            float dh = fminf(rv[3] * 0.2f, CLIPV);
            float pcx = dx * w + cx, pcy = dy * h + cy;
            float pw = __expf(dw) * w, ph = __expf(dh) * h;
            float x1 = fminf(fmaxf(pcx - 0.5f * pw, 0.f), IMGW);
            float y1 = fminf(fmaxf(pcy - 0.5f * ph, 0.f), IMGH);
            float x2 = fminf(fmaxf(pcx + 0.5f * pw, 0.f), IMGW);
            float y2 = fminf(fmaxf(pcy + 0.5f * ph, 0.f), IMGH);
            bool valid = (prob > SCORE_TH) && ((x2 - x1) >= MINSZ) && ((y2 - y1) >= MINSZ);
            krow[c - 1] = f2k(valid ? prob : NEGV);
        }
    }
}

// ---------------- init ----------------
__global__ void k_init(unsigned* hist, unsigned* prefix, unsigned* kcur,
                       unsigned* cntGT, unsigned* cntEQ,
                       unsigned* candKey, unsigned* candIdx) {
    int t = blockIdx.x * blockDim.x + threadIdx.x;
    if (t < B_ * 256) hist[t] = 0;
    if (t < B_) { prefix[t] = 0; kcur[t] = KPRE; cntGT[t] = 0; cntEQ[t] = 0; }
    if (t < B_ * KPRE) { candKey[t] = f2k(NEGV); candIdx[t] = 0; }
}

// ---------------- radix select: histogram pass ----------------
__global__ __launch_bounds__(256) void k_hist(const unsigned* __restrict__ keys,
                                              const unsigned* __restrict__ prefix,
                                              unsigned* __restrict__ hist, int pass) {
    __shared__ unsigned lh[256];
    int b = blockIdx.y;
    lh[threadIdx.x] = 0;
    __syncthreads();
    unsigned pfx = prefix[b];
    unsigned hm = (pass == 0) ? 0u : (0xFFFFFFFFu << (32 - 8 * pass));
    int shift = 24 - 8 * pass;
    const unsigned* kb = keys + (size_t)b * M_;
    for (int i = blockIdx.x * blockDim.x + threadIdx.x; i < M_; i += gridDim.x * blockDim.x) {
        unsigned k = kb[i];
        if ((k & hm) == pfx) atomicAdd(&lh[(k >> shift) & 0xFF], 1u);
    }
    __syncthreads();
    unsigned v = lh[threadIdx.x];
    if (v) atomicAdd(&hist[b * 256 + threadIdx.x], v);
}

// ---------------- radix select: pick byte ----------------
__global__ __launch_bounds__(256) void k_select(unsigned* __restrict__ hist,
                                                unsigned* __restrict__ prefix,
                                                unsigned* __restrict__ kcur, int pass) {
    __shared__ unsigned lh[256];
    int b = blockIdx.x, t = threadIdx.x;
    lh[t] = hist[b * 256 + t];
    hist[b * 256 + t] = 0;     // reset for next pass
    __syncthreads();
    if (t == 0) {
        unsigned k = kcur[b], total = 0;
        int chosen = 0;
        for (int v = 255; v >= 0; --v) {
            unsigned c = lh[v];
            if (total + c >= k) { chosen = v; k -= total; break; }
            total += c;
        }
        prefix[b] |= ((unsigned)chosen) << (24 - 8 * pass);
        kcur[b] = k;
    }
}

// ---------------- compaction ----------------
__global__ __launch_bounds__(256) void k_compact_gt(const unsigned* __restrict__ keys,
                                                    const unsigned* __restrict__ prefix,
                                                    unsigned* __restrict__ cntGT,
                                                    unsigned* __restrict__ candKey,
                                                    unsigned* __restrict__ candIdx) {
    int b = blockIdx.y;
    unsigned T = prefix[b];
    const unsigned* kb = keys + (size_t)b * M_;
    for (int i = blockIdx.x * blockDim.x + threadIdx.x; i < M_; i += gridDim.x * blockDim.x) {
        unsigned k = kb[i];
        if (k > T) {
            unsigned pos = atomicAdd(&cntGT[b], 1u);
            if (pos < KPRE) { candKey[b * KPRE + pos] = k; candIdx[b * KPRE + pos] = (unsigned)i; }
        }
    }
}

__global__ __launch_bounds__(256) void k_compact_eq(const unsigned* __restrict__ keys,
                                                    const unsigned* __restrict__ prefix,
                                                    const unsigned* __restrict__ cntGT,
                                                    unsigned* __restrict__ cntEQ,
                                                    unsigned* __restrict__ candKey,
                                                    unsigned* __restrict__ candIdx) {
    int b = blockIdx.y;
    unsigned T = prefix[b];
    unsigned base = cntGT[b];
    const unsigned* kb = keys + (size_t)b * M_;
    for (int i = blockIdx.x * blockDim.x + threadIdx.x; i < M_; i += gridDim.x * blockDim.x) {
        if (kb[i] == T) {
            unsigned pos = base + atomicAdd(&cntEQ[b], 1u);
            if (pos < KPRE) { candKey[b * KPRE + pos] = T; candIdx[b * KPRE + pos] = (unsigned)i; }
        }
    }
}

// ---------------- NMS: one block per image, whole working set in LDS -------
// dynamic-LDS layout (no static __shared__ in this kernel so the numeric LDS
// base for the async copies is __builtin_amdgcn_groupstaticsize() == 0)
#define L_SIDX 0u
#define L_SKEY 16384u
#define L_BX1  32768u
#define L_BY1  49152u
#define L_BX2  65536u
#define L_BY2  81920u
#define L_SSC  98304u
#define L_SLB  114688u
#define L_RV   131072u
#define L_RI   131200u
#define L_KI   131328u
#define L_KS   131728u
#define L_KO   132128u
#define L_PICK 132528u
#define L_PBOX 132544u
#define L_PLBL 132560u
#define NMS_SMEM 132608u

__global__ __launch_bounds__(1024) void k_nms(const float* __restrict__ rel,
                                              const float* __restrict__ props,
                                              const unsigned* __restrict__ candKey,
                                              const unsigned* __restrict__ candIdx,
                                              float* __restrict__ out) {
    extern __shared__ __align__(16) unsigned char dsm[];
    unsigned* sIdx = (unsigned*)(dsm + L_SIDX);
    unsigned* sKey = (unsigned*)(dsm + L_SKEY);
    float* bx1 = (float*)(dsm + L_BX1);
    float* by1 = (float*)(dsm + L_BY1);
    float* bx2 = (float*)(dsm + L_BX2);
    float* by2 = (float*)(dsm + L_BY2);
    float* sSc = (float*)(dsm + L_SSC);
    int*   sLb = (int*)(dsm + L_SLB);
    float* rv  = (float*)(dsm + L_RV);
    int*   ri  = (int*)(dsm + L_RI);
    int*   kI  = (int*)(dsm + L_KI);
    float* kS  = (float*)(dsm + L_KS);
    int*   kO  = (int*)(dsm + L_KO);
    int*   pick = (int*)(dsm + L_PICK);
    float* pbox = (float*)(dsm + L_PBOX);
    int*   plbl = (int*)(dsm + L_PLBL);

    const int t = threadIdx.x;
    const int b = blockIdx.x;

    // --- stage candidate idx/key straight into LDS (async, ASYNCcnt) ---
    {
        unsigned gss = __builtin_amdgcn_groupstaticsize();   // == 0 here
        unsigned long long idxBase = (unsigned long long)(uintptr_t)(candIdx + (size_t)b * KPRE);
        unsigned long long keyBase = (unsigned long long)(uintptr_t)(candKey + (size_t)b * KPRE);
        unsigned off16 = (unsigned)t * 16u;                  // 1024 threads x 16B = 16KB/array
        async_g2l_b128(gss + L_SIDX + off16, off16, idxBase);
        async_g2l_b128(gss + L_SKEY + off16, off16, keyBase);
        wait_async0();
    }
    __syncthreads();

    // --- decode the 4096 candidate boxes into LDS (gathers hit L2) ---
    const float* relB = rel + (size_t)b * N_ * C_ * 4;
    const float* prB  = props + (size_t)b * N_ * 4;
    for (int j = t; j < KPRE; j += 1024) {
        unsigned idx = sIdx[j];
        float sc = k2f(sKey[j]);
        int n = (int)(idx / CM1);
        int c = (int)(idx % CM1) + 1;
        const float* pr = prB + (size_t)n * 4;
        float p0 = pr[0], p1 = pr[1], p2 = pr[2], p3 = pr[3];
        float w = p2 - p0, h = p3 - p1;
        float cx = p0 + 0.5f * w, cy = p1 + 0.5f * h;
        const float* rr = relB + ((size_t)n * C_ + (size_t)c) * 4;
        float dx = rr[0] * 0.1f, dy = rr[1] * 0.1f;
        float dw = fminf(rr[2] * 0.2f, CLIPV), dh = fminf(rr[3] * 0.2f, CLIPV);
        float pcx = dx * w + cx, pcy = dy * h + cy;
        float pw = __expf(dw) * w, ph = __expf(dh) * h;
        bx1[j] = fminf(fmaxf(pcx - 0.5f * pw, 0.f), IMGW);
        by1[j] = fminf(fmaxf(pcy - 0.5f * ph, 0.f), IMGH);
        bx2[j] = fminf(fmaxf(pcx + 0.5f * pw, 0.f), IMGW);
        by2[j] = fminf(fmaxf(pcy + 0.5f * ph, 0.f), IMGH);
        sSc[j] = sc;
        sLb[j] = c;
    }
    __syncthreads();

    // --- greedy class-aware NMS, 100 picks ---
    const int lane = t & 31, wid = t >> 5;
    for (int it = 0; it < DET; ++it) {
        float bv = -3.4e38f; int bi = KPRE;
        for (int j = t; j < KPRE; j += 1024) {
            float v = sSc[j];
            if (v > bv || (v == bv && j < bi)) { bv = v; bi = j; }
        }
        for (int o = 16; o; o >>= 1) {                 // wave32 argmax
            float ov = __shfl_xor(bv, o, 32);
            int   oi = __shfl_xor(bi, o, 32);
            if (ov > bv || (ov == bv && oi < bi)) { bv = ov; bi = oi; }
        }
        if (lane == 0) { rv[wid] = bv; ri[wid] = bi; }
        __syncthreads();
        if (t == 0) {
            float bb = rv[0]; int bj = ri[0];
            for (int wv = 1; wv < 32; ++wv) {
                float ov = rv[wv]; int oi = ri[wv];
                if (ov > bb || (ov == bb && oi < bj)) { bb = ov; bj = oi; }
            }
            kI[it] = bj; kS[it] = bb; kO[it] = (bb > -5e8f) ? 1 : 0;
            pick[0] = bj;
            pbox[0] = bx1[bj]; pbox[1] = by1[bj]; pbox[2] = bx2[bj]; pbox[3] = by2[bj];
            plbl[0] = sLb[bj];
        }
        __syncthreads();
        int pi = pick[0];
        float px1 = pbox[0], py1 = pbox[1], px2 = pbox[2], py2 = pbox[3];
        int pl = plbl[0];
        float pa = (px2 - px1) * (py2 - py1);
        for (int j = t; j < KPRE; j += 1024) {
            float ix1 = fmaxf(px1, bx1[j]);
            float iy1 = fmaxf(py1, by1[j]);
            float ix2 = fminf(px2, bx2[j]);
            float iy2 = fminf(py2, by2[j]);
            float iw = fmaxf(ix2 - ix1, 0.f), ih = fmaxf(iy2 - iy1, 0.f);
            float inter = iw * ih;
            float a2 = (bx2[j] - bx1[j]) * (by2[j] - by1[j]);
            float iou = inter / fmaxf(pa + a2 - inter, 1e-8f);
            if (j == pi || (iou > NMS_TH && sLb[j] == pl)) sSc[j] = NEGV;
        }
        __syncthreads();
    }

    // --- emit: boxes [B,100,4], scores [B,100], labels [B,100] (as float) ---
    if (t < DET) {
        int ok = kO[t]; int i = kI[t];
        float* ob = out + ((size_t)b * DET + t) * 4;
        ob[0] = ok ? bx1[i] : 0.f;
        ob[1] = ok ? by1[i] : 0.f;
        ob[2] = ok ? bx2[i] : 0.f;
        ob[3] = ok ? by2[i] : 0.f;
        out[(size_t)B_ * DET * 4 + (size_t)b * DET + t] = ok ? kS[t] : 0.f;
        out[(size_t)B_ * DET * 5 + (size_t)b * DET + t] = ok ? (float)sLb[i] : -1.f;
    }
}

// ---------------- host-side launch ----------------
extern "C" void kernel_launch(void* const* d_in, const int* in_sizes, int n_in,
                              void* d_out, int out_size, void* d_ws, size_t ws_size,
                              hipStream_t stream) {
    (void)in_sizes; (void)n_in; (void)out_size; (void)ws_size;
    const float* logits = (const float*)d_in[0];
    const float* rel    = (const float*)d_in[1];
    const float* props  = (const float*)d_in[2];
    float* out = (float*)d_out;

    char* ws = (char*)d_ws;
    unsigned* keys    = (unsigned*)(ws + OFF_KEYS);
    unsigned* hist    = (unsigned*)(ws + OFF_HIST);
    unsigned* prefix  = (unsigned*)(ws + OFF_PREFIX);
    unsigned* kcur    = (unsigned*)(ws + OFF_KCUR);
    unsigned* cntGT   = (unsigned*)(ws + OFF_CNTGT);
    unsigned* cntEQ   = (unsigned*)(ws + OFF_CNTEQ);
    unsigned* candKey = (unsigned*)(ws + OFF_CKEY);
    unsigned* candIdx = (unsigned*)(ws + OFF_CIDX);

    k_init<<<dim3((B_ * KPRE + 255) / 256), 256, 0, stream>>>(hist, prefix, kcur,
                                                              cntGT, cntEQ, candKey, candIdx);
    k_scores<<<dim3((B_ * N_) / 8), 256, 0, stream>>>(logits, rel, props, keys);
    for (int p = 0; p < 4; ++p) {
        k_hist<<<dim3(64, B_), 256, 0, stream>>>(keys, prefix, hist, p);
        k_select<<<dim3(B_), 256, 0, stream>>>(hist, prefix, kcur, p);
    }
    k_compact_gt<<<dim3(64, B_), 256, 0, stream>>>(keys, prefix, cntGT, candKey, candIdx);
    k_compact_eq<<<dim3(64, B_), 256, 0, stream>>>(keys, prefix, cntGT, cntEQ, candKey, candIdx);
    k_nms<<<dim3(B_), 1024, NMS_SMEM, stream>>>(rel, props, candKey, candIdx, out);
}